// GAT_77790447665814
// MI455X (gfx1250) — compile-verified
//
#include <hip/hip_runtime.h>
#include <hip/hip_bf16.h>

#define N_NODES   100000
#define N_EDGES   600000
#define NUM_GRAPHS 512
#define IN_DIM    128
#define HID       64
#define ETOT      (N_EDGES + N_NODES)   // edges + self loops

typedef float v2f __attribute__((ext_vector_type(2)));
typedef float v8f __attribute__((ext_vector_type(8)));
typedef unsigned int u32x4 __attribute__((ext_vector_type(4)));
typedef int i32x4 __attribute__((ext_vector_type(4)));
typedef int i32x8 __attribute__((ext_vector_type(8)));

// ---------- order-preserving float<->uint map for atomic max ----------
__device__ __forceinline__ unsigned fmap(float f) {
  unsigned u = __float_as_uint(f);
  return (u & 0x80000000u) ? ~u : (u | 0x80000000u);
}
__device__ __forceinline__ float funmap(unsigned u) {
  unsigned b = (u & 0x80000000u) ? (u & 0x7fffffffu) : ~u;
  return __uint_as_float(b);
}

// ---------- TDM: DMA a [rows x cols] fp32 row-major matrix into LDS ----------
// Descriptor per CDNA5 ISA ch.8 (2-D tensor, tile == whole tensor, no padding,
// no gather, no multicast). Issued by one wave; completion via TENSORcnt.
__device__ __forceinline__ void tdm_load_2d_f32(const float* gptr, unsigned lds_off,
                                                int rows, int cols) {
  unsigned long long ga = (unsigned long long)(const void*)gptr;
  u32x4 g0;
  g0[0] = 1u;                                   // count=1, is_restore=0, gather=0
  g0[1] = lds_off;                              // lds_addr (bytes)
  g0[2] = (unsigned)ga;                         // global_addr[31:0]
  g0[3] = (unsigned)((ga >> 32) & 0x1FFFFFFu)   // global_addr[56:32]
          | (2u << 30);                         // type=2 ("image")
  i32x8 g1;
  g1[0] = 2 << 16;                              // data_size=2 -> 4 bytes
  g1[1] = (cols & 0xffff) << 16;                // tensor_dim0[15:0]
  g1[2] = ((unsigned)cols >> 16)                // tensor_dim0[31:16]
          | ((rows & 0xffff) << 16);            // tensor_dim1[15:0]
  g1[3] = ((unsigned)rows >> 16)                // tensor_dim1[31:16]
          | ((cols & 0xffff) << 16);            // tile_dim0
  g1[4] = rows & 0xffff;                        // tile_dim1 (tile_dim2=0)
  g1[5] = cols;                                 // tensor_dim0_stride[31:0]
  g1[6] = 0;                                    // stride hi / dim1_stride lo
  g1[7] = 0;                                    // dim1_stride hi
  i32x4 gz4 = {0, 0, 0, 0};                     // groups 2/3 unused (2-D tensor)
  i32x8 gz8 = {0, 0, 0, 0, 0, 0, 0, 0};
  __builtin_amdgcn_tensor_load_to_lds(g0, g1, gz4, gz4, gz8, 0);
}

// ---------- fp32 WMMA GEMM: C[M,N] = A[M,K] @ B[K,N], M multiple of 16 ----------
// B staged in LDS by the Tensor Data Mover; one wave per 16x16 output tile;
// blockDim.x = (N/16)*32, gridDim.x = M/16
template<int K, int N>
__global__ void __launch_bounds__(256)
gat_wmma_gemm(const float* __restrict__ A, const float* __restrict__ B,
              float* __restrict__ C) {
  __shared__ float sB[K * N];
  if (threadIdx.x < 32) {                         // wave 0 drives the TDM
    tdm_load_2d_f32(B, (unsigned)(unsigned long long)(void*)sB, K, N);
    __builtin_amdgcn_s_wait_tensorcnt(0);
  }
  __syncthreads();                                // publish LDS to all waves

  const int wave = threadIdx.x >> 5;
  const int lane = threadIdx.x & 31;
  const int half = lane >> 4;      // 0: K pair {0,1}; 1: K pair {2,3}
  const int l    = lane & 15;      // A: row within tile, B: col within tile
  const int m0   = blockIdx.x << 4;
  const int n0   = wave << 4;

  v8f acc = {0.f, 0.f, 0.f, 0.f, 0.f, 0.f, 0.f, 0.f};
  const float* arow = A + (size_t)(m0 + l) * K + 2 * half;
  const float* brow = sB + 2 * half * N + n0 + l;

#pragma unroll
  for (int k = 0; k < K; k += 4) {
    v2f a, b;
    a.x = arow[k];
    a.y = arow[k + 1];
    b.x = brow[k * N];
    b.y = brow[k * N + N];
    acc = __builtin_amdgcn_wmma_f32_16x16x4_f32(false, a, false, b,
                                                (short)0, acc, false, false);
  }
  // C/D layout: VGPR r -> row (r + 8*half), col = l
  float* crow = C + (size_t)(m0 + 8 * half) * N + n0 + l;
#pragma unroll
  for (int r = 0; r < 8; ++r)
    crow[(size_t)r * N] = acc[r];
}

// ---------- per-(node,head) attention logit projections ----------
__global__ void gat_alpha(const float* __restrict__ h, const float* __restrict__ a_s,
                          const float* __restrict__ a_d, float* __restrict__ as_out,
                          float* __restrict__ ad_out, int total, int H) {
  int idx = blockIdx.x * blockDim.x + threadIdx.x;
  if (idx >= total) return;
  int hd = idx % H;
  const float* hp = h + (size_t)idx * HID;   // h is [N, H, HID] row-major -> row idx
  const float* vs = a_s + hd * HID;
  const float* vd = a_d + hd * HID;
  float ss = 0.f, sd = 0.f;
#pragma unroll 8
  for (int c = 0; c < HID; ++c) {
    float v = hp[c];
    ss = fmaf(v, vs[c], ss);
    sd = fmaf(v, vd[c], sd);
  }
  as_out[idx] = ss;
  ad_out[idx] = sd;
}

// ---------- edge pass A: leaky-relu logits + segment max ----------
__global__ void gat_edge_max(const int* __restrict__ ei, const float* __restrict__ as,
                             const float* __restrict__ ad, float* __restrict__ e_buf,
                             unsigned* __restrict__ mmax, int H) {
  int idx = blockIdx.x * blockDim.x + threadIdx.x;
  int total = ETOT * H;
  if (idx >= total) return;
  int edge = idx / H, hd = idx - edge * H;
  int src, dst;
  if (edge < N_EDGES) { src = ei[edge]; dst = ei[N_EDGES + edge]; }
  else                { src = dst = edge - N_EDGES; }
  float e = as[src * H + hd] + ad[dst * H + hd];
  e = (e >= 0.f) ? e : 0.2f * e;             // leaky relu, slope 0.2
  e_buf[idx] = e;
  atomicMax(&mmax[dst * H + hd], fmap(e));
}

// ---------- edge pass B: exp(e - m) + segment sum ----------
__global__ void gat_edge_exp(const int* __restrict__ ei, float* __restrict__ e_buf,
                             const unsigned* __restrict__ mmax,
                             float* __restrict__ denom, int H) {
  int idx = blockIdx.x * blockDim.x + threadIdx.x;
  int total = ETOT * H;
  if (idx >= total) return;
  int edge = idx / H, hd = idx - edge * H;
  int dst = (edge < N_EDGES) ? ei[N_EDGES + edge] : edge - N_EDGES;
  float m = funmap(mmax[dst * H + hd]);
  if (!__builtin_isfinite(m)) m = 0.f;       // matches jnp.where(isfinite(m), m, 0)
  float ex = __expf(e_buf[idx] - m);
  e_buf[idx] = ex;
  atomicAdd(&denom[dst * H + hd], ex);
}

// ---------- edge pass C: weighted gather-scatter, float4 per thread ----------
__global__ void gat_edge_scatter(const int* __restrict__ ei, const float* __restrict__ h,
                                 const float* __restrict__ e_buf,
                                 const float* __restrict__ denom,
                                 float* __restrict__ out, int H) {
  const int F = H * HID;
  const int T = F / 4;                        // float4 chunks per edge
  long long t = (long long)blockIdx.x * blockDim.x + threadIdx.x;
  long long total = (long long)ETOT * T;
  if (t >= total) return;
  int edge = (int)(t / T);
  int cq = (int)(t % T) * 4;
  int src, dst;
  if (edge < N_EDGES) { src = ei[edge]; dst = ei[N_EDGES + edge]; }
  else                { src = dst = edge - N_EDGES; }
  int hd = cq / HID;
  float alpha = e_buf[edge * H + hd] / fmaxf(denom[dst * H + hd], 1e-16f);
  const float4 hv = *reinterpret_cast<const float4*>(h + (size_t)src * F + cq);
  float* op = out + (size_t)dst * F + cq;
  atomicAdd(op + 0, alpha * hv.x);
  atomicAdd(op + 1, alpha * hv.y);
  atomicAdd(op + 2, alpha * hv.z);
  atomicAdd(op + 3, alpha * hv.w);
}

// ---------- bias (+ optional relu) ----------
__global__ void gat_bias_act(float* __restrict__ x, const float* __restrict__ b,
                             int total, int F, int relu) {
  int idx = blockIdx.x * blockDim.x + threadIdx.x;
  if (idx >= total) return;
  float v = x[idx] + b[idx % F];
  x[idx] = relu ? fmaxf(v, 0.f) : v;
}

// ---------- per-graph sum / max / count ----------
__global__ void gat_pool(const float* __restrict__ h, const int* __restrict__ batch,
                         float* __restrict__ psum, unsigned* __restrict__ pmax,
                         float* __restrict__ pcnt) {
  int idx = blockIdx.x * blockDim.x + threadIdx.x;
  int total = N_NODES * HID;
  if (idx >= total) return;
  int n = idx / HID, c = idx - n * HID;
  int g = batch[n];
  float v = h[idx];
  atomicAdd(&psum[g * HID + c], v);
  atomicMax(&pmax[g * HID + c], fmap(v));
  if (c == 0) atomicAdd(&pcnt[g], 1.0f);
}

// ---------- final MLP: one 64-thread block per graph ----------
__global__ void __launch_bounds__(HID)
gat_mlp(const float* __restrict__ psum, const unsigned* __restrict__ pmax,
        const float* __restrict__ pcnt, const float* __restrict__ Wf1,
        const float* __restrict__ bf1, const float* __restrict__ Wf2,
        const float* __restrict__ bf2, float* __restrict__ out) {
  __shared__ float pooled[2 * HID];
  __shared__ float red[HID];
  int g = blockIdx.x, j = threadIdx.x;
  float cnt = pcnt[g];
  float inv = 1.f / fmaxf(cnt, 1.f);
  pooled[j] = psum[g * HID + j] * inv;                       // mean
  pooled[HID + j] = (cnt > 0.f) ? funmap(pmax[g * HID + j]) : 0.f;  // max
  __syncthreads();
  float acc = bf1[j];
#pragma unroll 8
  for (int k = 0; k < 2 * HID; ++k)
    acc = fmaf(pooled[k], Wf1[k * HID + j], acc);
  acc = fmaxf(acc, 0.f);
  red[j] = acc * Wf2[j];
  __syncthreads();
  for (int s = HID / 2; s > 0; s >>= 1) {
    if (j < s) red[j] += red[j + s];
    __syncthreads();
  }
  if (j == 0) out[g] = red[0] + bf2[0];
}

static inline unsigned cdiv(long long a, long long b) {
  return (unsigned)((a + b - 1) / b);
}

extern "C" void kernel_launch(void* const* d_in, const int* in_sizes, int n_in,
                              void* d_out, int out_size, void* d_ws, size_t ws_size,
                              hipStream_t stream) {
  const float* x      = (const float*)d_in[0];
  const int*   ei     = (const int*)d_in[1];    // [2, 600000]
  const int*   batch  = (const int*)d_in[2];
  const float* W1     = (const float*)d_in[3];
  const float* a_src1 = (const float*)d_in[4];
  const float* a_dst1 = (const float*)d_in[5];
  const float* b1     = (const float*)d_in[6];
  const float* W2     = (const float*)d_in[7];
  const float* a_src2 = (const float*)d_in[8];
  const float* a_dst2 = (const float*)d_in[9];
  const float* b2     = (const float*)d_in[10];
  const float* Wf1    = (const float*)d_in[11];
  const float* bf1    = (const float*)d_in[12];
  const float* Wf2    = (const float*)d_in[13];
  const float* bf2    = (const float*)d_in[14];
  float* out = (float*)d_out;

  // ---- workspace layout (floats) ----
  float* ws = (float*)d_ws;
  float*    h1   = ws;                      // 12,800,000  (reused as out2 later)
  float*    out1 = h1 + 12800000;           // 12,800,000
  float*    h2   = out1 + 12800000;         //  6,400,000
  float*    as1  = h2 + 6400000;            //    200,000
  float*    ad1  = as1 + 200000;            //    200,000
  unsigned* m1   = (unsigned*)(ad1 + 200000);  // 200,000
  float*    den1 = (float*)m1 + 200000;     //    200,000
  float*    e1   = den1 + 200000;           //  1,400,000
  float*    as2  = e1 + 1400000;            //    100,000
  float*    ad2  = as2 + 100000;            //    100,000
  unsigned* m2   = (unsigned*)(ad2 + 100000);  // 100,000
  float*    den2 = (float*)m2 + 100000;     //    100,000
  float*    e2   = den2 + 100000;           //    700,000
  float*    psum = e2 + 700000;             //     32,768
  unsigned* pmax = (unsigned*)(psum + 32768);  // 32,768
  float*    pcnt = (float*)pmax + 32768;    //        512
  float*    out2 = h1;                      // alias: h1 is dead after layer-1 scatter

  const int BS = 256;

  // ===== Layer 1 =====
  gat_wmma_gemm<IN_DIM, 2 * HID><<<N_NODES / 16, 256, 0, stream>>>(x, W1, h1);
  gat_alpha<<<cdiv(N_NODES * 2, BS), BS, 0, stream>>>(h1, a_src1, a_dst1, as1, ad1,
                                                      N_NODES * 2, 2);
  (void)hipMemsetAsync(out1, 0, (size_t)12800000 * 4, stream);
  (void)hipMemsetAsync(m1, 0, (size_t)200000 * 4, stream);
  (void)hipMemsetAsync(den1, 0, (size_t)200000 * 4, stream);
  gat_edge_max<<<cdiv((long long)ETOT * 2, BS), BS, 0, stream>>>(ei, as1, ad1, e1, m1, 2);
  gat_edge_exp<<<cdiv((long long)ETOT * 2, BS), BS, 0, stream>>>(ei, e1, m1, den1, 2);
  gat_edge_scatter<<<cdiv((long long)ETOT * 32, BS), BS, 0, stream>>>(ei, h1, e1, den1,
                                                                     out1, 2);
  gat_bias_act<<<cdiv(N_NODES * 128, BS), BS, 0, stream>>>(out1, b1, N_NODES * 128,
                                                           128, 1);

  // ===== Layer 2 =====
  gat_wmma_gemm<2 * HID, HID><<<N_NODES / 16, 128, 0, stream>>>(out1, W2, h2);
  gat_alpha<<<cdiv(N_NODES, BS), BS, 0, stream>>>(h2, a_src2, a_dst2, as2, ad2,
                                                  N_NODES, 1);
  (void)hipMemsetAsync(out2, 0, (size_t)6400000 * 4, stream);  // h1 region, dead now
  (void)hipMemsetAsync(m2, 0, (size_t)100000 * 4, stream);
  (void)hipMemsetAsync(den2, 0, (size_t)100000 * 4, stream);
  gat_edge_max<<<cdiv((long long)ETOT, BS), BS, 0, stream>>>(ei, as2, ad2, e2, m2, 1);
  gat_edge_exp<<<cdiv((long long)ETOT, BS), BS, 0, stream>>>(ei, e2, m2, den2, 1);
  gat_edge_scatter<<<cdiv((long long)ETOT * 16, BS), BS, 0, stream>>>(ei, h2, e2, den2,
                                                                     out2, 1);
  gat_bias_act<<<cdiv(N_NODES * HID, BS), BS, 0, stream>>>(out2, b2, N_NODES * HID,
                                                           HID, 0);

  // ===== Pool + MLP =====
  (void)hipMemsetAsync(psum, 0, (size_t)NUM_GRAPHS * HID * 4, stream);
  (void)hipMemsetAsync(pmax, 0, (size_t)NUM_GRAPHS * HID * 4, stream);
  (void)hipMemsetAsync(pcnt, 0, (size_t)NUM_GRAPHS * 4, stream);
  gat_pool<<<cdiv((long long)N_NODES * HID, BS), BS, 0, stream>>>(out2, batch, psum,
                                                                  pmax, pcnt);
  gat_mlp<<<NUM_GRAPHS, HID, 0, stream>>>(psum, pmax, pcnt, Wf1, bf1, Wf2, bf2, out);
  (void)in_sizes; (void)n_in; (void)out_size; (void)ws_size;
}